// SpatialGridProjector_50379966382936
// MI455X (gfx1250) — compile-verified
//
#include <hip/hip_runtime.h>

typedef __attribute__((ext_vector_type(2))) float v2f;
typedef __attribute__((ext_vector_type(8))) float v8f;

#define LDIM 128
#define BDIM 4
#define QDIM 8

// Trilinear sample with zero padding outside [0,127]^3, matching the
// reference: per-axis validity masks fold into the weights, addresses clamped.
__device__ __forceinline__ float trilerp(const float* __restrict__ vol,
                                         float x, float y, float z) {
  float xf = floorf(x), yf = floorf(y), zf = floorf(z);
  float fx = x - xf, fy = y - yf, fz = z - zf;
  int x0 = (int)xf, y0 = (int)yf, z0 = (int)zf;
  int x1 = x0 + 1, y1 = y0 + 1, z1 = z0 + 1;
  float wx0 = (x0 >= 0 && x0 < LDIM) ? (1.0f - fx) : 0.0f;
  float wx1 = (x1 >= 0 && x1 < LDIM) ? fx : 0.0f;
  float wy0 = (y0 >= 0 && y0 < LDIM) ? (1.0f - fy) : 0.0f;
  float wy1 = (y1 >= 0 && y1 < LDIM) ? fy : 0.0f;
  float wz0 = (z0 >= 0 && z0 < LDIM) ? (1.0f - fz) : 0.0f;
  float wz1 = (z1 >= 0 && z1 < LDIM) ? fz : 0.0f;
  int cx0 = min(max(x0, 0), LDIM - 1), cx1 = min(max(x1, 0), LDIM - 1);
  int cy0 = min(max(y0, 0), LDIM - 1), cy1 = min(max(y1, 0), LDIM - 1);
  int cz0 = min(max(z0, 0), LDIM - 1), cz1 = min(max(z1, 0), LDIM - 1);
  int p00 = (cz0 * LDIM + cy0) * LDIM;
  int p01 = (cz0 * LDIM + cy1) * LDIM;
  int p10 = (cz1 * LDIM + cy0) * LDIM;
  int p11 = (cz1 * LDIM + cy1) * LDIM;
  float r = vol[p00 + cx0] * (wx0 * wy0 * wz0);
  r = fmaf(vol[p00 + cx1], wx1 * wy0 * wz0, r);
  r = fmaf(vol[p01 + cx0], wx0 * wy1 * wz0, r);
  r = fmaf(vol[p01 + cx1], wx1 * wy1 * wz0, r);
  r = fmaf(vol[p10 + cx0], wx0 * wy0 * wz1, r);
  r = fmaf(vol[p10 + cx1], wx1 * wy0 * wz1, r);
  r = fmaf(vol[p11 + cx0], wx0 * wy1 * wz1, r);
  r = fmaf(vol[p11 + cx1], wx1 * wy1 * wz1, r);
  return r;
}

// Block = 256 threads = 8 wave32. Waves 0..3 -> tile 2*blockIdx, waves 4..7 ->
// tile 2*blockIdx+1; within a tile each wave integrates a 32-step k-slice.
// Per k step, three V_WMMA_F32_16X16X4_F32 produce the unnormalized x/y/z
// sample coordinates for the whole 16x16 pixel tile (scale+offset folded in).
__global__ __launch_bounds__(256) void
SpatialGridProjector_kernel(const float* __restrict__ density,
                            const float* __restrict__ rotation,
                            float* __restrict__ out) {
  __shared__ float redbuf[8][256];

  const int lane = threadIdx.x & 31;
  const int wave = threadIdx.x >> 5;
  const int tile = blockIdx.x * 2 + (wave >> 2);   // 0..2047
  const int kslice = wave & 3;                     // k in [32*kslice, 32*kslice+32)

  const int img = tile >> 6;                       // b*Q + q  (64 tiles/image)
  const int t = tile & 63;
  const int jt = (t >> 3) << 4;                    // tile row base (j)
  const int it = (t & 7) << 4;                     // tile col base (i)
  const int b = img / QDIM;

  const float* __restrict__ vol = density + (size_t)b * (LDIM * LDIM * LDIM);
  const float* __restrict__ R = rotation + (size_t)img * 9;

  const float s = 0.5f * (float)(LDIM - 1);        // 63.5
  const float step = 2.0f / (float)(LDIM - 1);

  const float R00 = R[0], R01 = R[1], R02 = R[2];
  const float R10 = R[3], R11 = R[4], R12 = R[5];
  const float R20 = R[6], R21 = R[7], R22 = R[8];

  const bool lo = (lane < 16);
  const int n = lane & 15;
  const float lj = -1.0f + step * (float)(jt + n); // A row coordinate (M = j)
  const float li = -1.0f + step * (float)(it + n); // B column coordinate (N = i)

  // A (16x4 f32): lanes 0-15 hold K=0,1 for M=lane; lanes 16-31 hold K=2,3 (=0).
  v2f A;
  A.x = lo ? lj : 0.0f;     // K=0 : lin[j]
  A.y = lo ? 1.0f : 0.0f;   // K=1 : 1

  // B (4x16 f32), per output component m:
  //   K=0 row: s*R[1][m]          (pairs with lin[j])
  //   K=1 row: s*li*R[0][m] + s + s*lin[k]*R[2][m]   (pairs with 1)
  const float bx0 = lo ? s * R10 : 0.0f;
  const float by0 = lo ? s * R11 : 0.0f;
  const float bz0 = lo ? s * R12 : 0.0f;
  const float baseX = lo ? fmaf(s * li, R00, s) : 0.0f;
  const float baseY = lo ? fmaf(s * li, R01, s) : 0.0f;
  const float baseZ = lo ? fmaf(s * li, R02, s) : 0.0f;
  const float kcX = lo ? s * R20 : 0.0f;
  const float kcY = lo ? s * R21 : 0.0f;
  const float kcZ = lo ? s * R22 : 0.0f;

  v8f acc = {};
  const v8f czero = {};
  const int k0 = kslice * (LDIM / 4);
  for (int k = k0; k < k0 + (LDIM / 4); ++k) {
    const float lk = -1.0f + step * (float)k;
    v2f Bx, By, Bz;
    Bx.x = bx0; Bx.y = fmaf(lk, kcX, baseX);
    By.x = by0; By.y = fmaf(lk, kcY, baseY);
    Bz.x = bz0; Bz.y = fmaf(lk, kcZ, baseZ);
    // D = A*B + 0 : unnormalized sample coords for all 256 tile pixels.
    v8f xg = __builtin_amdgcn_wmma_f32_16x16x4_f32(false, A, false, Bx,
                                                   (short)0, czero, false, false);
    v8f yg = __builtin_amdgcn_wmma_f32_16x16x4_f32(false, A, false, By,
                                                   (short)0, czero, false, false);
    v8f zg = __builtin_amdgcn_wmma_f32_16x16x4_f32(false, A, false, Bz,
                                                   (short)0, czero, false, false);
#pragma unroll
    for (int r = 0; r < 8; ++r) {
      acc[r] += trilerp(vol, xg[r], yg[r], zg[r]);
    }
  }

  // Cross-wave k-slice reduction through LDS (deterministic).
#pragma unroll
  for (int r = 0; r < 8; ++r) redbuf[wave][r * 32 + lane] = acc[r];
  __syncthreads();

  if (kslice == 0) {
    const int mBase = lo ? 0 : 8;  // D layout: VGPR r -> M = r (+8 for hi lanes)
#pragma unroll
    for (int r = 0; r < 8; ++r) {
      const int e = r * 32 + lane;
      float v = redbuf[wave][e] + redbuf[wave + 1][e] +
                redbuf[wave + 2][e] + redbuf[wave + 3][e];
      const int jj = jt + mBase + r;   // output row    (j)
      const int ii = it + n;           // output column (i) -> coalesced
      out[((size_t)img * LDIM + jj) * LDIM + ii] = v;
    }
  }
}

extern "C" void kernel_launch(void* const* d_in, const int* in_sizes, int n_in,
                              void* d_out, int out_size, void* d_ws, size_t ws_size,
                              hipStream_t stream) {
  (void)in_sizes; (void)n_in; (void)d_ws; (void)ws_size; (void)out_size;
  const float* density  = (const float*)d_in[0];   // (B, D, D, D) f32
  const float* rotation = (const float*)d_in[1];   // (B, Q, 3, 3) f32
  float* out = (float*)d_out;                      // (B, Q, L, L) f32

  // 2048 tiles, 2 tiles per block (8 waves: 2 tiles x 4 k-slices).
  const int nblocks = (BDIM * QDIM * (LDIM / 16) * (LDIM / 16)) / 2;  // 1024
  SpatialGridProjector_kernel<<<nblocks, 256, 0, stream>>>(density, rotation, out);
}